// QuantumRegression_65481071403789
// MI455X (gfx1250) — compile-verified
//
#include <hip/hip_runtime.h>
#include <hip/hip_bf16.h>

// ---------------------------------------------------------------------------
// 12-qubit statevector simulator, one workgroup per batch element.
// State (4096 complex64 = 32KB) lives entirely in LDS.
// Fused 16x16 complex unitaries (4-qubit groups) applied with
// V_WMMA_F32_16X16X4_F32 (fp32 precision preserved).
// ---------------------------------------------------------------------------

typedef __attribute__((ext_vector_type(2))) float v2f;
typedef __attribute__((ext_vector_type(8))) float v8f;

#define NQ   12
#define NL   4
#define DIM  4096           // 2^NQ
#define NT   256            // threads per block (8 waves)
#define APT  (DIM / NT)     // 16 amplitudes per thread

// D(16x16,f32) = A(16x4,f32) x B(4x16,f32) + C
// args: (neg_a, A, neg_b, B, c_mod, C, reuse_a, reuse_b); f32 has no A/B neg.
__device__ __forceinline__ v8f wmma4(v2f a, v2f b, v8f c) {
  return __builtin_amdgcn_wmma_f32_16x16x4_f32(false, a, false, b, (short)0, c,
                                               false, false);
}

// Fragment index helpers (ISA 16x16x4 f32 layouts):
//   A: lane holds A[lane&15][k0], A[lane&15][k0+1],  k0 = 2*(lane>>4)  (+4*slice)
//   B: lane holds B[k0][lane&15], B[k0+1][lane&15]   (mirror of A)
//   C/D: vgpr v holds D[v + 8*(lane>>4)][lane&15]
//
// Each apply_* runs 8 waves x 2 tiles = 16 disjoint 16x16x16 complex matmuls.
// The tile "loop" is a compile-time 2-iteration unroll so control flow stays
// uniform and EXEC is all-ones at every WMMA (ISA requirement) with no
// divergent-loop EXEC masking around the matrix ops.

// --- group 0: low 4 bits.  newS = S @ U^T  (A = state rows, B = U^T) -------
__device__ __forceinline__ void apply_low(float* sRe, float* sIm,
                                          const float* gRe, const float* gIm) {
  const int lane = threadIdx.x & 31;
  const int wv   = threadIdx.x >> 5;
  const int mn   = lane & 15;
  const int kh   = (lane >> 4) * 2;
#pragma unroll
  for (int rep = 0; rep < 2; ++rep) {
    const int base = (wv + rep * 8) * 256;
    v8f dre = {}, dim = {};
#pragma unroll
    for (int s = 0; s < 4; ++s) {
      const int k0 = s * 4 + kh;
      v2f ar, ai, br, bi, nbi;
      ar.x = sRe[base + mn * 16 + k0];     ar.y = sRe[base + mn * 16 + k0 + 1];
      ai.x = sIm[base + mn * 16 + k0];     ai.y = sIm[base + mn * 16 + k0 + 1];
      // B = U^T: B[k][n] = U[n][k]
      br.x = gRe[mn * 16 + k0];            br.y = gRe[mn * 16 + k0 + 1];
      bi.x = gIm[mn * 16 + k0];            bi.y = gIm[mn * 16 + k0 + 1];
      nbi.x = -bi.x;                       nbi.y = -bi.y;
      dre = wmma4(ar, br, dre);   dre = wmma4(ai, nbi, dre);
      dim = wmma4(ar, bi, dim);   dim = wmma4(ai, br, dim);
    }
    const int mb = (lane >> 4) * 8;
#pragma unroll
    for (int v = 0; v < 8; ++v) {
      sRe[base + (mb + v) * 16 + mn] = dre[v];
      sIm[base + (mb + v) * 16 + mn] = dim[v];
    }
  }
}

// --- group 1: middle 4 bits.  newS[a,j,b] = sum_k U[j,k] S[a,k,b] ----------
__device__ __forceinline__ void apply_mid(float* sRe, float* sIm,
                                          const float* gRe, const float* gIm) {
  const int lane = threadIdx.x & 31;
  const int wv   = threadIdx.x >> 5;
  const int mn   = lane & 15;
  const int kh   = (lane >> 4) * 2;
#pragma unroll
  for (int rep = 0; rep < 2; ++rep) {
    const int base = (wv + rep * 8) * 256;
    v8f dre = {}, dim = {};
#pragma unroll
    for (int s = 0; s < 4; ++s) {
      const int k0 = s * 4 + kh;
      v2f ar, ai, br, bi, nbi;
      ar.x = gRe[mn * 16 + k0];            ar.y = gRe[mn * 16 + k0 + 1];
      ai.x = gIm[mn * 16 + k0];            ai.y = gIm[mn * 16 + k0 + 1];
      br.x = sRe[base + k0 * 16 + mn];     br.y = sRe[base + (k0 + 1) * 16 + mn];
      bi.x = sIm[base + k0 * 16 + mn];     bi.y = sIm[base + (k0 + 1) * 16 + mn];
      nbi.x = -bi.x;                       nbi.y = -bi.y;
      dre = wmma4(ar, br, dre);   dre = wmma4(ai, nbi, dre);
      dim = wmma4(ar, bi, dim);   dim = wmma4(ai, br, dim);
    }
    const int mb = (lane >> 4) * 8;
#pragma unroll
    for (int v = 0; v < 8; ++v) {
      sRe[base + (mb + v) * 16 + mn] = dre[v];
      sIm[base + (mb + v) * 16 + mn] = dim[v];
    }
  }
}

// --- group 2: high 4 bits.  newS[j,m] = sum_k U[j,k] S[k,m] ----------------
__device__ __forceinline__ void apply_high(float* sRe, float* sIm,
                                           const float* gRe, const float* gIm) {
  const int lane = threadIdx.x & 31;
  const int wv   = threadIdx.x >> 5;
  const int mn   = lane & 15;
  const int kh   = (lane >> 4) * 2;
#pragma unroll
  for (int rep = 0; rep < 2; ++rep) {
    const int cb = (wv + rep * 8) * 16;
    v8f dre = {}, dim = {};
#pragma unroll
    for (int s = 0; s < 4; ++s) {
      const int k0 = s * 4 + kh;
      v2f ar, ai, br, bi, nbi;
      ar.x = gRe[mn * 16 + k0];            ar.y = gRe[mn * 16 + k0 + 1];
      ai.x = gIm[mn * 16 + k0];            ai.y = gIm[mn * 16 + k0 + 1];
      br.x = sRe[k0 * 256 + cb + mn];      br.y = sRe[(k0 + 1) * 256 + cb + mn];
      bi.x = sIm[k0 * 256 + cb + mn];      bi.y = sIm[(k0 + 1) * 256 + cb + mn];
      nbi.x = -bi.x;                       nbi.y = -bi.y;
      dre = wmma4(ar, br, dre);   dre = wmma4(ai, nbi, dre);
      dim = wmma4(ar, bi, dim);   dim = wmma4(ai, br, dim);
    }
    const int mb = (lane >> 4) * 8;
#pragma unroll
    for (int v = 0; v < 8; ++v) {
      sRe[(mb + v) * 256 + cb + mn] = dre[v];
      sIm[(mb + v) * 256 + cb + mn] = dim[v];
    }
  }
}

// ---------------------------------------------------------------------------
// Pre-kernel: fuse the batch-shared Rot gates of layer l, qubit-group g into
// 16x16 complex matrices.  ws layout: re at [(l*3+g)*256 + e], im at +3072.
// Bit position of qubit q in the flat amplitude index is (11 - q).
// ---------------------------------------------------------------------------
__global__ __launch_bounds__(NT) void rot_fuse_kernel(
    const float* __restrict__ w, float* __restrict__ ws) {
  const int l = blockIdx.x / 3;
  const int g = blockIdx.x % 3;
  const int e = threadIdx.x;
  const int j = e >> 4, k = e & 15;
  float pr = 1.f, pi = 0.f;
#pragma unroll
  for (int bt = 0; bt < 4; ++bt) {
    const int q  = 11 - (4 * g + bt);
    const int jt = (j >> bt) & 1, kt = (k >> bt) & 1;
    const float phi = w[(l * NQ + q) * 3 + 0];
    const float th  = w[(l * NQ + q) * 3 + 1];
    const float om  = w[(l * NQ + q) * 3 + 2];
    const float ct = cosf(0.5f * th), st = sinf(0.5f * th);
    const float a = 0.5f * (phi + om), bb = 0.5f * (phi - om);
    const float ca = cosf(a), sa = sinf(a), cb = cosf(bb), sb = sinf(bb);
    float ur, ui;
    if (jt == 0 && kt == 0)      { ur =  ct * ca; ui = -ct * sa; }  // e^{-ia} ct
    else if (jt == 0 && kt == 1) { ur = -st * cb; ui = -st * sb; }  // -e^{ib} st
    else if (jt == 1 && kt == 0) { ur =  st * cb; ui = -st * sb; }  // e^{-ib} st
    else                         { ur =  ct * ca; ui =  ct * sa; }  // e^{ia} ct
    const float nr = pr * ur - pi * ui;
    pi = pr * ui + pi * ur;
    pr = nr;
  }
  ws[(l * 3 + g) * 256 + e]        = pr;
  ws[3072 + (l * 3 + g) * 256 + e] = pi;
}

// ---------------------------------------------------------------------------
// Main kernel: one block per batch element.
// ---------------------------------------------------------------------------
__global__ __launch_bounds__(NT) void qsim_kernel(
    const float* __restrict__ x, const float* __restrict__ wsG,
    const float* __restrict__ head_w, const float* __restrict__ head_b,
    float* __restrict__ out) {
  __shared__ float sRe[DIM], sIm[DIM];
  __shared__ float gRe[3 * 256], gIm[3 * 256];
  __shared__ float red[NQ * NT];

  const int b = blockIdx.x;
  const int t = threadIdx.x;

  __builtin_prefetch(wsG, 0, 1);  // global_prefetch_b8 of shared gate tables

  // |0...0>
#pragma unroll
  for (int i = 0; i < APT; ++i) {
    const int y = t + NT * i;
    sRe[y] = (y == 0) ? 1.f : 0.f;
    sIm[y] = 0.f;
  }

  // Per-batch fused RX embedding matrices (group g covers bits 4g..4g+3).
#pragma unroll
  for (int g = 0; g < 3; ++g) {
    const int j = t >> 4, k = t & 15;
    float pr = 1.f, pi = 0.f;
#pragma unroll
    for (int bt = 0; bt < 4; ++bt) {
      const int q = 11 - (4 * g + bt);
      const float ang = 0.5f * x[b * NQ + q];
      const float c = cosf(ang), s = sinf(ang);
      const int jt = (j >> bt) & 1, kt = (k >> bt) & 1;
      const float ur = (jt == kt) ? c : 0.f;
      const float ui = (jt == kt) ? 0.f : -s;
      const float nr = pr * ur - pi * ui;
      pi = pr * ui + pi * ur;
      pr = nr;
    }
    gRe[g * 256 + t] = pr;
    gIm[g * 256 + t] = pi;
  }
  __syncthreads();

  apply_low (sRe, sIm, gRe,       gIm);       __syncthreads();
  apply_mid (sRe, sIm, gRe + 256, gIm + 256); __syncthreads();
  apply_high(sRe, sIm, gRe + 512, gIm + 512); __syncthreads();

  for (int l = 0; l < NL; ++l) {
    // load this layer's fused Rot matrices (shared across batch, L2-hot)
#pragma unroll
    for (int g = 0; g < 3; ++g) {
      gRe[g * 256 + t] = wsG[(l * 3 + g) * 256 + t];
      gIm[g * 256 + t] = wsG[3072 + (l * 3 + g) * 256 + t];
    }
    __syncthreads();
    apply_low (sRe, sIm, gRe,       gIm);       __syncthreads();
    apply_mid (sRe, sIm, gRe + 256, gIm + 256); __syncthreads();
    apply_high(sRe, sIm, gRe + 512, gIm + 512); __syncthreads();

    // CNOT ring, range r = l+1: composed GF(2) permutation, gather form.
    const int r = l + 1;
    float rr[APT], ri[APT];
#pragma unroll
    for (int i = 0; i < APT; ++i) {
      const int y = t + NT * i;
      int xx = y;
#pragma unroll
      for (int q = NQ - 1; q >= 0; --q) {
        const int cbit = 11 - q;
        const int tbit = 11 - ((q + r) % NQ);
        xx ^= ((xx >> cbit) & 1) << tbit;
      }
      rr[i] = sRe[xx];
      ri[i] = sIm[xx];
    }
    __syncthreads();
#pragma unroll
    for (int i = 0; i < APT; ++i) {
      const int y = t + NT * i;
      sRe[y] = rr[i];
      sIm[y] = ri[i];
    }
    __syncthreads();
  }

  // <Z_q> (bit 11-q) + linear head; deterministic reduction.
  float acc[NQ];
#pragma unroll
  for (int q = 0; q < NQ; ++q) acc[q] = 0.f;
#pragma unroll
  for (int i = 0; i < APT; ++i) {
    const int y = t + NT * i;
    const float p = sRe[y] * sRe[y] + sIm[y] * sIm[y];
#pragma unroll
    for (int q = 0; q < NQ; ++q)
      acc[q] += ((y >> (11 - q)) & 1) ? -p : p;
  }
#pragma unroll
  for (int q = 0; q < NQ; ++q) red[q * NT + t] = acc[q];
  __syncthreads();
  if (t < NQ) {
    float ssum = 0.f;
    for (int i = 0; i < NT; ++i) ssum += red[t * NT + i];
    red[t * NT] = ssum;
  }
  __syncthreads();
  if (t == 0) {
    float o = head_b[0];
#pragma unroll
    for (int q = 0; q < NQ; ++q) o += head_w[q] * red[q * NT];
    out[b] = o;
  }
}

extern "C" void kernel_launch(void* const* d_in, const int* in_sizes, int n_in,
                              void* d_out, int out_size, void* d_ws,
                              size_t ws_size, hipStream_t stream) {
  const float* x  = (const float*)d_in[0];  // (B, 12)
  const float* w  = (const float*)d_in[1];  // (4, 12, 3)
  const float* hw = (const float*)d_in[2];  // (1, 12)
  const float* hb = (const float*)d_in[3];  // (1,)
  float* out = (float*)d_out;
  float* ws  = (float*)d_ws;                // needs 12*256*2 floats = 24 KB
  const int B = in_sizes[0] / NQ;

  rot_fuse_kernel<<<NL * 3, NT, 0, stream>>>(w, ws);
  qsim_kernel<<<B, NT, 0, stream>>>(x, ws, hw, hb, out);
}